// ProposalLayer_5317169512504
// MI455X (gfx1250) — compile-verified
//
#include <hip/hip_runtime.h>
#include <hip/hip_bf16.h>

// ---------------- problem constants ----------------
#define BATCH       2
#define NANCH       261888
#define PRE_NMS     6000
#define BOXN        6016           // PRE_NMS padded to multiple of 32
#define PROP        1000
#define NMS_THR     0.7f
#define CAND_CAP    8192
#define NBINS       65536
#define WORDS       188            // BOXN/32 == ceil(6000/32)
#define TILES       188            // BOXN/32 tiles of 32 boxes

typedef float v2f __attribute__((ext_vector_type(2)));
typedef float v8f __attribute__((ext_vector_type(8)));
typedef unsigned long long u64;
typedef unsigned int u32;

// ---------------- workspace layout (u32 indices) ----------------
#define HIST_W      (BATCH * NBINS)
#define OFF_THR     (HIST_W)
#define OFF_CNT     (OFF_THR + BATCH)
#define OFF_CAND    (OFF_CNT + BATCH)              // 131076 (×4B -> 8B aligned)
#define CAND_W      (BATCH * CAND_CAP * 2)         // u64 as 2×u32
#define OFF_IX      (OFF_CAND + CAND_W)
#define IX_W        (BATCH * PRE_NMS)
#define OFF_BOX     (OFF_IX + IX_W)                // 175844, 16B aligned
#define BOX_W       (BATCH * BOXN * 4)             // padded boxes (pad stays zero)
#define OFF_MASK    (OFF_BOX + BOX_W)
#define MASK_W      (BATCH * PRE_NMS * WORDS)
#define WS_TOTAL    (OFF_MASK + MASK_W)            // ~2.48M u32 (~9.9 MB)

// ---------------- 0: zero scratch ----------------
__global__ void zero_ws_kernel(u32* __restrict__ ws, long long n) {
    long long i = (long long)blockIdx.x * blockDim.x + threadIdx.x;
    long long stride = (long long)gridDim.x * blockDim.x;
    for (; i < n; i += stride) ws[i] = 0u;
}

// ---------------- 1: 16-bit radix histogram of scores ----------------
__global__ void hist_kernel(const float* __restrict__ rpn_class, u32* __restrict__ hist) {
    int gid = blockIdx.x * blockDim.x + threadIdx.x;
    if (gid >= BATCH * NANCH) return;
    int b = gid / NANCH;
    __builtin_prefetch((const void*)(rpn_class + (long long)gid * 2 + 4096), 0, 0);
    float s = rpn_class[(long long)gid * 2 + 1];   // class-1 score
    u32 key = __float_as_uint(s);                  // positive float: order-preserving
    atomicAdd(&hist[b * NBINS + (key >> 16)], 1u);
}

// ---------------- 2: find threshold bin (6000th largest) ----------------
__global__ __launch_bounds__(1024)
void thresh_kernel(const u32* __restrict__ hist, u32* __restrict__ thrT) {
    __shared__ u32 partial[1024];
    int b = blockIdx.x;
    const u32* h = hist + b * NBINS;
    u32 sum = 0;
    #pragma unroll 4
    for (int i = 0; i < 64; ++i) sum += h[threadIdx.x * 64 + i];
    partial[threadIdx.x] = sum;
    __syncthreads();
    if (threadIdx.x == 0) {
        u32 acc = 0; int seg = 0;
        for (int t = 1023; t >= 0; --t) {
            if (acc + partial[t] >= PRE_NMS) { seg = t; break; }
            acc += partial[t];
        }
        u32 T = (u32)(seg * 64);
        for (int i = 63; i >= 0; --i) {
            acc += h[seg * 64 + i];
            if (acc >= PRE_NMS) { T = (u32)(seg * 64 + i); break; }
        }
        thrT[b] = T;
    }
}

// ---------------- 3: compact candidates >= threshold ----------------
__global__ void compact_kernel(const float* __restrict__ rpn_class,
                               const u32* __restrict__ thrT,
                               u32* __restrict__ counter,
                               u64* __restrict__ cand) {
    int gid = blockIdx.x * blockDim.x + threadIdx.x;
    if (gid >= BATCH * NANCH) return;
    int b = gid / NANCH;
    int n = gid - b * NANCH;
    u32 key = __float_as_uint(rpn_class[(long long)gid * 2 + 1]);
    if ((key >> 16) >= thrT[b]) {
        u32 pos = atomicAdd(&counter[b], 1u);
        if (pos < CAND_CAP)   // composite: key desc, then index asc (via ~n)
            cand[b * CAND_CAP + pos] = ((u64)key << 32) | (u32)(~(u32)n);
    }
}

// ---------------- 4: LDS bitonic sort (8192 u64, one WGP) ----------------
__global__ __launch_bounds__(1024)
void sort_kernel(const u64* __restrict__ cand, u32* __restrict__ ixbuf) {
    __shared__ u64 s[CAND_CAP];     // 64 KB of the 320 KB/WGP LDS
    int b = blockIdx.x;
    for (int i = threadIdx.x; i < CAND_CAP; i += 1024) s[i] = cand[b * CAND_CAP + i];
    __syncthreads();
    for (u32 k = 2; k <= CAND_CAP; k <<= 1) {
        for (u32 j = k >> 1; j > 0; j >>= 1) {
            for (u32 i = threadIdx.x; i < CAND_CAP; i += 1024) {
                u32 l = i ^ j;
                if (l > i) {
                    bool up = ((i & k) == 0);
                    u64 a = s[i], c = s[l];
                    bool swp = up ? (a < c) : (a > c);   // descending overall
                    if (swp) { s[i] = c; s[l] = a; }
                }
            }
            __syncthreads();
        }
    }
    for (int i = threadIdx.x; i < PRE_NMS; i += 1024)
        ixbuf[b * PRE_NMS + i] = ~(u32)(s[i] & 0xFFFFFFFFull);
}

// ---------------- 5: gather + box decode + clip ----------------
__global__ void gather_kernel(const float* __restrict__ rpn_bbox,
                              const float* __restrict__ anchors,
                              const u32* __restrict__ ixbuf,
                              float* __restrict__ boxes) {
    int gid = blockIdx.x * blockDim.x + threadIdx.x;
    if (gid >= BATCH * PRE_NMS) return;
    int b = gid / PRE_NMS;
    int i = gid - b * PRE_NMS;
    u32 n = ixbuf[gid];
    const float* a = anchors  + ((long long)b * NANCH + n) * 4;
    const float* d = rpn_bbox + ((long long)b * NANCH + n) * 4;
    float y1 = a[0], x1 = a[1], y2 = a[2], x2 = a[3];
    float h = y2 - y1, w = x2 - x1;
    float dy = d[0] * 0.1f, dx = d[1] * 0.1f, dh = d[2] * 0.2f, dw = d[3] * 0.2f;
    float cy = y1 + 0.5f * h + dy * h;
    float cx = x1 + 0.5f * w + dx * w;
    h *= expf(dh);  w *= expf(dw);
    float ny1 = cy - 0.5f * h, nx1 = cx - 0.5f * w;
    float ny2 = ny1 + h,       nx2 = nx1 + w;
    float* o = boxes + ((long long)b * BOXN + i) * 4;
    o[0] = fminf(fmaxf(ny1, 0.f), 1.f);
    o[1] = fminf(fmaxf(nx1, 0.f), 1.f);
    o[2] = fminf(fmaxf(ny2, 0.f), 1.f);
    o[3] = fminf(fmaxf(nx2, 0.f), 1.f);
}

// ---------------- 6: IoU suppression bitmask, 32x32 tile / wave, 4 WMMAs --------
// The matmul-expressible part of IoU (area_i + area_j) is computed as rank-2
// outer products on the matrix pipe: v_wmma_f32_16x16x4_f32 with
//   A row M = (area_i[M], 1, 0, 0),  B col N = (1, area_j[N], 0, 0).
// A 32-column tile covers exactly one 32-bit mask word, so each (row, word)
// is produced by exactly one wave -> plain stores, no atomics.
__global__ __launch_bounds__(32)
void iou_kernel(const float* __restrict__ boxes, u32* __restrict__ mask) {
    int jb = blockIdx.x, ib = blockIdx.y, b = blockIdx.z;
    if (jb > ib) return;                    // lower-triangular tiles only
    __shared__ float4 rbS[32], cbS[32];
    int l = threadIdx.x;
    const float4* bb = (const float4*)(boxes + (long long)b * BOXN * 4);
    rbS[l] = bb[ib * 32 + l];               // pad boxes are zero-area (no suppress)
    cbS[l] = bb[jb * 32 + l];
    __syncthreads();

    int m = l & 15;
    bool lo = (l < 16);
    float4 c0 = cbS[m], c1 = cbS[16 + m];
    float4 r0 = rbS[m], r1 = rbS[16 + m];
    float ar0 = (r0.z - r0.x) * (r0.w - r0.y);
    float ar1 = (r1.z - r1.x) * (r1.w - r1.y);
    float ac0 = (c0.z - c0.x) * (c0.w - c0.y);
    float ac1 = (c1.z - c1.x) * (c1.w - c1.y);

    // 16x4 f32 A-layout: lanes 0-15 hold K=0,1; lanes 16-31 hold K=2,3 (zeros)
    v2f a0, a1, b0, b1;
    a0.x = lo ? ar0 : 0.f;  a0.y = lo ? 1.f : 0.f;
    a1.x = lo ? ar1 : 0.f;  a1.y = lo ? 1.f : 0.f;
    b0.x = lo ? 1.f : 0.f;  b0.y = lo ? ac0 : 0.f;
    b1.x = lo ? 1.f : 0.f;  b1.y = lo ? ac1 : 0.f;
    v8f z = {};
    v8f dm[4];
    dm[0] = __builtin_amdgcn_wmma_f32_16x16x4_f32(false, a0, false, b0, (short)0, z, false, false);
    dm[1] = __builtin_amdgcn_wmma_f32_16x16x4_f32(false, a0, false, b1, (short)0, z, false, false);
    dm[2] = __builtin_amdgcn_wmma_f32_16x16x4_f32(false, a1, false, b0, (short)0, z, false, false);
    dm[3] = __builtin_amdgcn_wmma_f32_16x16x4_f32(false, a1, false, b1, (short)0, z, false, false);

    u32* mrow = mask + (long long)b * PRE_NMS * WORDS;
    #pragma unroll
    for (int ti = 0; ti < 2; ++ti) {
        #pragma unroll
        for (int rr = 0; rr < 8; ++rr) {
            // C/D layout: VGPR rr, lanes 0-15 -> row rr, lanes 16-31 -> row rr+8
            float4 bi = rbS[ti * 16 + (lo ? rr : rr + 8)];
            u32 bits_lo = 0u, bits_hi = 0u;   // rows ti*16+rr and ti*16+rr+8
            #pragma unroll
            for (int tj = 0; tj < 2; ++tj) {
                float4 bj = (tj == 0) ? c0 : c1;
                float ih = fmaxf(fminf(bi.z, bj.z) - fmaxf(bi.x, bj.x), 0.f);
                float iw = fmaxf(fminf(bi.w, bj.w) - fmaxf(bi.y, bj.y), 0.f);
                float inter = ih * iw;
                float denom = dm[ti * 2 + tj][rr] - inter + 1e-8f;  // WMMA area sum
                bool sup = inter > NMS_THR * denom;                 // iou > thr
                u32 bal = __builtin_amdgcn_ballot_w32(sup);
                bits_lo |= (bal & 0xFFFFu) << (tj * 16);
                bits_hi |= (bal >> 16)     << (tj * 16);
            }
            int gi_lo = ib * 32 + ti * 16 + rr;
            int gi_hi = gi_lo + 8;
            if (l == 0 && gi_lo < PRE_NMS)
                mrow[(long long)gi_lo * WORDS + jb] = bits_lo;
            if (l == 1 && gi_hi < PRE_NMS)
                mrow[(long long)gi_hi * WORDS + jb] = bits_hi;
        }
    }
}

// ---------------- 7: serial greedy scan, one wave32 per batch ----------------
// keep-mask (188 words) lives in 6 registers/lane; kept boxes stream to d_out.
__global__ __launch_bounds__(32)
void nms_kernel(const float* __restrict__ boxes, const u32* __restrict__ mask,
                float* __restrict__ out) {
    int b = blockIdx.x;
    int t = threadIdx.x;
    u32 kept[6] = {0u, 0u, 0u, 0u, 0u, 0u};
    const u32* mbase = mask + (long long)b * PRE_NMS * WORDS;
    const float4* bb = (const float4*)(boxes + (long long)b * BOXN * 4);
    float4* ob = (float4*)out + b * PROP;
    int cnt = 0;
    for (int i = 0; i < PRE_NMS; ++i) {
        const u32* row = mbase + (long long)i * WORDS;
        u32 sup = 0u;
        #pragma unroll
        for (int s = 0; s < 6; ++s) {
            int w = s * 32 + t;
            if (w < WORDS) sup |= row[w] & kept[s];
        }
        bool any = (__builtin_amdgcn_ballot_w32(sup != 0u) != 0u);
        if (!any) {
            int W = i >> 5;
            if (t == (W & 31)) kept[W >> 5] |= (1u << (i & 31));
            if (cnt < PROP) {
                if (t == 0) ob[cnt] = bb[i];
                ++cnt;                     // uniform across the wave
            }
        }
    }
    for (int slot = cnt + t; slot < PROP; slot += 32)
        ob[slot] = make_float4(0.f, 0.f, 0.f, 0.f);
}

// ---------------- launcher ----------------
extern "C" void kernel_launch(void* const* d_in, const int* in_sizes, int n_in,
                              void* d_out, int out_size, void* d_ws, size_t ws_size,
                              hipStream_t stream) {
    const float* rpn_class = (const float*)d_in[0];   // (B,N,2)
    const float* rpn_bbox  = (const float*)d_in[1];   // (B,N,4)
    const float* anchors   = (const float*)d_in[2];   // (B,N,4)
    float* out = (float*)d_out;                       // (B,1000,4)

    u32*  ws     = (u32*)d_ws;
    u32*  hist   = ws;
    u32*  thrT   = ws + OFF_THR;
    u32*  cnt    = ws + OFF_CNT;
    u64*  cand   = (u64*)(ws + OFF_CAND);
    u32*  ixbuf  = ws + OFF_IX;
    float* boxes = (float*)(ws + OFF_BOX);
    u32*  mask   = ws + OFF_MASK;

    // 0: zero hist/counters/candidates/box-pad/mask (deterministic per call)
    zero_ws_kernel<<<4096, 256, 0, stream>>>(ws, (long long)WS_TOTAL);

    // 1: score histogram
    int nElem = BATCH * NANCH;
    hist_kernel<<<(nElem + 255) / 256, 256, 0, stream>>>(rpn_class, hist);

    // 2: threshold bin per batch
    thresh_kernel<<<BATCH, 1024, 0, stream>>>(hist, thrT);

    // 3: compact candidates
    compact_kernel<<<(nElem + 255) / 256, 256, 0, stream>>>(rpn_class, thrT, cnt, cand);

    // 4: in-LDS bitonic sort of 8192 composites per batch
    sort_kernel<<<BATCH, 1024, 0, stream>>>(cand, ixbuf);

    // 5: gather + decode + clip boxes
    int nBox = BATCH * PRE_NMS;
    gather_kernel<<<(nBox + 255) / 256, 256, 0, stream>>>(rpn_bbox, anchors, ixbuf, boxes);

    // 6: pairwise IoU suppression bitmask (4 WMMAs / wave), triangular 32x32 tiles
    dim3 iougrid(TILES, TILES, BATCH);
    iou_kernel<<<iougrid, 32, 0, stream>>>(boxes, mask);

    // 7: serial greedy keep-scan + output
    nms_kernel<<<BATCH, 32, 0, stream>>>(boxes, mask, out);

    (void)in_sizes; (void)n_in; (void)out_size; (void)ws_size;
}